// Reservoir_24927990186506
// MI455X (gfx1250) — compile-verified
//
#include <hip/hip_runtime.h>

#define T_STEPS 4096
#define N_RES   2048
#define IN_DIM  64
#define OUT_DIM 64

#define NWG           128
#define ROWS_PER_WG   (N_RES / NWG)          // 16
#define BLOCK         256                    // 8 waves (wave32)
#define WAVES         (BLOCK / 32)
#define ROWS_PER_WAVE (ROWS_PER_WG / WAVES)  // 2

#define LEAK        0.3f
#define NOISE_SCALE 0.01f

typedef float v2f __attribute__((ext_vector_type(2)));
typedef float v8f __attribute__((ext_vector_type(8)));

// ---------------------------------------------------------------------------
// Device-wide split barrier on an atomic counter (persistent-kernel sync).
// Counter only ever increases within one launch; init kernel resets it per
// launch so graph replays are deterministic.
// ---------------------------------------------------------------------------
__device__ __forceinline__ void grid_barrier(unsigned* counter, unsigned target) {
  __threadfence();                 // release this thread's stores to device scope
  __syncthreads();                 // all threads in WG done + fenced
  if (threadIdx.x == 0) {
    __hip_atomic_fetch_add(counter, 1u, __ATOMIC_RELEASE, __HIP_MEMORY_SCOPE_AGENT);
    while (__hip_atomic_load(counter, __ATOMIC_ACQUIRE, __HIP_MEMORY_SCOPE_AGENT) < target) {
      __builtin_amdgcn_s_sleep(2);
    }
  }
  __syncthreads();
  __threadfence();                 // acquire: invalidate stale near caches
}

__global__ void esn_init(unsigned* counter) {
  if (threadIdx.x == 0 && blockIdx.x == 0) *counter = 0u;
}

// ---------------------------------------------------------------------------
// Persistent recurrence kernel: each WG owns 16 rows of W. Per step:
// stage prev state (8KB) + u[t] into LDS, each wave computes 2 rows as a
// lane-strided b128-vectorized dot over L2-resident W, wave-reduce, apply
// tanh + leaky integration, publish to states[t+1], grid barrier.
// ---------------------------------------------------------------------------
__global__ void __launch_bounds__(BLOCK)
esn_recurrence(const float* __restrict__ u, const float* __restrict__ noise,
               const float* __restrict__ W_in, const float* __restrict__ W,
               float* __restrict__ states, unsigned* __restrict__ counter) {
  __shared__ float s_prev[N_RES];
  __shared__ float s_u[IN_DIM];

  const int tid  = threadIdx.x;
  const int lane = tid & 31;
  const int wave = tid >> 5;
  const int row0 = blockIdx.x * ROWS_PER_WG;

  // states[0][:] = 0 (initial reservoir state)
  for (int i = tid; i < ROWS_PER_WG; i += BLOCK) states[row0 + i] = 0.0f;
  unsigned target = NWG;
  grid_barrier(counter, target);

  for (int t = 0; t < T_STEPS; ++t) {
    const float* prev = states + (size_t)t * N_RES;
    for (int i = tid; i < N_RES; i += BLOCK) s_prev[i] = prev[i];
    if (tid < IN_DIM) s_u[tid] = u[(size_t)t * IN_DIM + tid];
    __syncthreads();

#pragma unroll
    for (int r = 0; r < ROWS_PER_WAVE; ++r) {
      const int row = row0 + wave * ROWS_PER_WAVE + r;
      const float4* w4 = reinterpret_cast<const float4*>(W + (size_t)row * N_RES);
      const float4* s4 = reinterpret_cast<const float4*>(s_prev);
      float acc = 0.0f;
#pragma unroll 4
      for (int i = lane; i < N_RES / 4; i += 32) {   // global_load_b128 stream of W
        const float4 wv = w4[i];
        const float4 sv = s4[i];
        acc = fmaf(wv.x, sv.x, acc);
        acc = fmaf(wv.y, sv.y, acc);
        acc = fmaf(wv.z, sv.z, acc);
        acc = fmaf(wv.w, sv.w, acc);
      }
      // fused input term: dot(u[t], W_in[row]) spread across lanes
      const float* wi = W_in + (size_t)row * IN_DIM;
      acc = fmaf(wi[lane], s_u[lane], acc);
      acc = fmaf(wi[lane + 32], s_u[lane + 32], acc);
      // wave32 tree reduction
#pragma unroll
      for (int off = 16; off > 0; off >>= 1) acc += __shfl_down(acc, off, 32);
      if (lane == 0) {
        const float it = acc + noise[(size_t)t * N_RES + row] * NOISE_SCALE;
        const float ns = tanhf(it);
        states[(size_t)(t + 1) * N_RES + row] =
            (1.0f - LEAK) * s_prev[row] + LEAK * ns;
      }
    }
    target += NWG;
    grid_barrier(counter, target);
  }
}

// ---------------------------------------------------------------------------
// Readout GEMM: out[T,64] = states[T,2048] @ r_w[64,2048]^T + r_b.
// fp32 WMMA 16x16x4, one wave per 16x16 output tile (block = 4 waves covers
// the full N=64). Per-lane A/B layout per ISA: K = 2*laneHi + {0,1}.
// ---------------------------------------------------------------------------
__global__ void __launch_bounds__(128)
esn_readout(const float* __restrict__ states, const float* __restrict__ r_w,
            const float* __restrict__ r_b, float* __restrict__ out) {
  const int tileM  = blockIdx.x;         // 0..255 (16 time steps each)
  const int wave   = threadIdx.x >> 5;   // tileN = wave, N tile = wave*16
  const int lane   = threadIdx.x & 31;
  const int laneLo = lane & 15;
  const int laneHi = lane >> 4;
  const int n      = wave * 16 + laneLo;

  // A row: state for time (tileM*16 + laneLo); states row t+1 holds step t.
  const float* arow = states + ((size_t)(tileM * 16 + laneLo) + 1) * N_RES;
  // B column n: r_w[n, :] (B = r_w^T so B[k][n] = r_w[n][k])
  const float* brow = r_w + (size_t)n * N_RES;

  v8f c = {};
  for (int k = 0; k < N_RES; k += 4) {
    const int kk = k + 2 * laneHi;
    v2f a, b;
    a.x = arow[kk];
    a.y = arow[kk + 1];
    b.x = brow[kk];
    b.y = brow[kk + 1];
    c = __builtin_amdgcn_wmma_f32_16x16x4_f32(
        /*neg_a=*/false, a, /*neg_b=*/false, b,
        /*c_mod=*/(short)0, c, /*reuse_a=*/false, /*reuse_b=*/false);
  }

  const float bias = r_b[n];
#pragma unroll
  for (int i = 0; i < 8; ++i) {
    const int m = tileM * 16 + i + 8 * laneHi;     // D vgpr i -> row i + 8*laneHi
    out[(size_t)m * OUT_DIM + n] = c[i] + bias;
  }
}

// ---------------------------------------------------------------------------
extern "C" void kernel_launch(void* const* d_in, const int* in_sizes, int n_in,
                              void* d_out, int out_size, void* d_ws, size_t ws_size,
                              hipStream_t stream) {
  const float* u     = (const float*)d_in[0];
  const float* noise = (const float*)d_in[1];
  const float* W_in  = (const float*)d_in[2];
  const float* W     = (const float*)d_in[3];
  const float* r_w   = (const float*)d_in[4];
  const float* r_b   = (const float*)d_in[5];
  float* out = (float*)d_out;

  unsigned* counter = (unsigned*)d_ws;
  float* states     = (float*)((char*)d_ws + 256);   // (T+1) x N_RES floats

  hipLaunchKernelGGL(esn_init, dim3(1), dim3(1), 0, stream, counter);
  hipLaunchKernelGGL(esn_recurrence, dim3(NWG), dim3(BLOCK), 0, stream,
                     u, noise, W_in, W, states, counter);
  hipLaunchKernelGGL(esn_readout, dim3(T_STEPS / 16), dim3(128), 0, stream,
                     states, r_w, r_b, out);
}